// MambaBlock_84980222918824
// MI455X (gfx1250) — compile-verified
//
#include <hip/hip_runtime.h>
#include <hip/hip_bf16.h>

// Mamba block for MI455X (gfx1250), wave32.
//  - Large GEMMs: LDS-tiled double-buffered fp32 WMMA (V_WMMA_F32_16X16X4_F32),
//    staged with async global->LDS (GLOBAL_LOAD_ASYNC_TO_LDS_B128 / ASYNCcnt).
//  - Small N=80 GEMM: direct WMMA kernel.
//  - Depthwise conv + SiLU, and a per-channel register selective scan.

#define B_SZ    2
#define SEQ_L   1024
#define DM      768      // D_MODEL
#define NS      16       // N_STATE
#define KC      4        // K_CONV
#define DR      48       // D_RANK
#define XP_W    (DR + 2*NS)   // 80

typedef __attribute__((ext_vector_type(2))) float v2f;
typedef __attribute__((ext_vector_type(8))) float v8f;

__device__ __forceinline__ float softplus_clip(float v) {
    v = (v > 20.0f) ? v : log1pf(__expf(v));
    return fminf(fmaxf(v, 1e-4f), 0.1f);
}

__device__ __forceinline__ void async_b128_to_lds(const float* g, const float* lds) {
    // GLOBAL_LOAD_ASYNC_TO_LDS_B128: VDST = LDS byte offset (low 32 bits of the
    // flat shared address), VADDR = 64-bit global address. Tracked by ASYNCcnt.
    const unsigned l = (unsigned)(uintptr_t)lds;
    asm volatile("global_load_async_to_lds_b128 %0, %1, off"
                 :: "v"(l), "v"((unsigned long long)(uintptr_t)g)
                 : "memory");
}

__device__ __forceinline__ void wait_async0() {
#if __has_builtin(__builtin_amdgcn_s_wait_asynccnt)
    __builtin_amdgcn_s_wait_asynccnt(0);
#else
    asm volatile("s_wait_asynccnt 0x0" ::: "memory");
#endif
}

// ---------------------------------------------------------------------------
// LDS-tiled WMMA GEMM: C[M,N] = A[M,K] * B[K,N] (+bias[N]) (+epilogue).
// Block = 256 threads (8 wave32) computes a 64(M) x 128(N) C block.
// Wave (wm 0..3, wn 0..1) computes a 16 x 64 strip = 4 accumulators, so each
// A fragment is reused by 4 WMMAs. K is consumed in chunks of 16, double
// buffered in LDS via async global->LDS b128 transfers.
// Requires: M%64==0, N%128==0, K%16==0.
// ---------------------------------------------------------------------------
#define BM  64
#define BN  128
#define BKC 16

__global__ __launch_bounds__(256)
void mamba_wmma_gemm_tiled(const float* __restrict__ A, int lda,
                           const float* __restrict__ B, int ldb,
                           float* __restrict__ C, int ldc,
                           int M, int N, int K,
                           const float* __restrict__ bias, int epilogue)
{
    __shared__ __align__(16) float sA[2][BM * BKC];   // 64 x 16, row-major
    __shared__ __align__(16) float sB[2][BKC * BN];   // 16 x 128, row-major

    const int tid  = threadIdx.x;
    const int lane = tid & 31;
    const int wv   = tid >> 5;       // 0..7
    const int wm   = wv & 3;         // m-tile within block (rows wm*16..)
    const int wn   = wv >> 2;        // n-half within block (cols wn*64..)
    const int mr   = lane & 15;
    const int h    = lane >> 4;

    const int nBlkN = N / BN;
    const int bm0 = (blockIdx.x / nBlkN) * BM;
    const int bn0 = (blockIdx.x % nBlkN) * BN;
    const int NC  = K / BKC;

    auto issue = [&](int c, int buf) {
        const int kc = c * BKC;
        {   // A tile: 64x16 = 1024 floats, one b128 per thread
            const int e = tid * 4;
            const int r = e >> 4, cc = e & 15;
            async_b128_to_lds(A + (long)(bm0 + r) * lda + kc + cc, &sA[buf][e]);
        }
#pragma unroll
        for (int i = 0; i < 2; ++i) {   // B tile: 16x128 = 2048 floats, two b128 per thread
            const int e = tid * 4 + i * 1024;
            const int r = e >> 7, cc = e & 127;
            async_b128_to_lds(B + (long)(kc + r) * ldb + bn0 + cc, &sB[buf][e]);
        }
    };

    v8f acc[4];
#pragma unroll
    for (int j = 0; j < 4; ++j) acc[j] = (v8f){};

    issue(0, 0);
    for (int c = 0; c < NC; ++c) {
        wait_async0();          // this wave's staged chunk is in LDS
        __syncthreads();        // everyone's chunk is in LDS; prev chunk fully consumed
        if (c + 1 < NC) issue(c + 1, (c + 1) & 1);

        const float* __restrict__ a_s = &sA[c & 1][0];
        const float* __restrict__ b_s = &sB[c & 1][0];
#pragma unroll
        for (int kk = 0; kk < BKC; kk += 4) {
            const int ar = (wm * 16 + mr) * BKC + kk + 2 * h;
            v2f a; a.x = a_s[ar]; a.y = a_s[ar + 1];          // ds_load_b64
            const int br = (kk + 2 * h) * BN + wn * 64 + mr;
#pragma unroll
            for (int j = 0; j < 4; ++j) {
                v2f b; b.x = b_s[br + j * 16]; b.y = b_s[br + BN + j * 16];
                acc[j] = __builtin_amdgcn_wmma_f32_16x16x4_f32(
                             false, a, false, b, (short)0, acc[j], false, false);
            }
        }
        __syncthreads();        // all reads of this buffer done before next overwrite cycle
    }

#pragma unroll
    for (int j = 0; j < 4; ++j) {
        const int col = bn0 + wn * 64 + j * 16 + mr;
        const float bb = bias ? bias[col] : 0.0f;
#pragma unroll
        for (int v = 0; v < 8; ++v) {
            float val = acc[j][v] + bb;
            if (epilogue == 1) val = softplus_clip(val);
            C[(long)(bm0 + wm * 16 + v + 8 * h) * ldc + col] = val;
        }
    }
}

// ---------------------------------------------------------------------------
// Direct WMMA GEMM (one wave per 16x16 tile) for odd shapes (here N=80).
// Requires M%16==0, N%16==0, K%4==0.
// ---------------------------------------------------------------------------
__global__ void mamba_wmma_gemm_f32(const float* __restrict__ A, int lda,
                                    const float* __restrict__ B, int ldb,
                                    float* __restrict__ C, int ldc,
                                    int M, int N, int K,
                                    const float* __restrict__ bias, int epilogue)
{
    const int lane  = threadIdx.x & 31;
    const int wave  = blockIdx.x * (blockDim.x >> 5) + (threadIdx.x >> 5);
    const int tilesM = M >> 4;
    const int tilesN = N >> 4;
    if (wave >= tilesM * tilesN) return;     // wave-uniform exit

    const int tm = wave / tilesN;
    const int tn = wave % tilesN;
    const int mr = lane & 15;
    const int h  = lane >> 4;

    const float* __restrict__ ap = A + (long)(tm * 16 + mr) * lda + 2 * h;
    const float* __restrict__ bp = B + (long)(2 * h) * ldb + tn * 16 + mr;
    const long bstep = 4l * ldb;

    v8f acc = {};
    for (int k = 0; k < K; k += 4) {
        v2f av; av.x = ap[0]; av.y = ap[1];
        v2f bv; bv.x = bp[0]; bv.y = bp[ldb];
        acc = __builtin_amdgcn_wmma_f32_16x16x4_f32(
                  false, av, false, bv, (short)0, acc, false, false);
        ap += 4;
        bp += bstep;
    }

    const int col = tn * 16 + mr;
    const float bv = bias ? bias[col] : 0.0f;
#pragma unroll
    for (int v = 0; v < 8; ++v) {
        float val = acc[v] + bv;
        if (epilogue == 1) val = softplus_clip(val);
        C[(long)(tm * 16 + v + 8 * h) * ldc + col] = val;
    }
}

// ---------------------------------------------------------------------------
// Depthwise causal conv (K=4) + bias + SiLU on first half of xz.
// ---------------------------------------------------------------------------
__global__ void mamba_conv_silu(const float* __restrict__ xz,
                                const float* __restrict__ cw,
                                const float* __restrict__ cb,
                                float* __restrict__ xs_act)
{
    const int idx = blockIdx.x * blockDim.x + threadIdx.x;   // over B*L*DM
    if (idx >= B_SZ * SEQ_L * DM) return;
    const int d = idx % DM;
    const int l = (idx / DM) % SEQ_L;
    const int b = idx / (DM * SEQ_L);

    float acc = cb[d];
#pragma unroll
    for (int k = 0; k < KC; ++k) {
        const int ls = l - (KC - 1) + k;
        if (ls >= 0)
            acc += cw[d * KC + k] * xz[((long)(b * SEQ_L + ls)) * (2 * DM) + d];
    }
    xs_act[idx] = acc / (1.0f + __expf(-acc));   // SiLU
}

// ---------------------------------------------------------------------------
// Selective scan fused with +u*D and silu(res) gate.
// grid = (DM/256, B), block = 256: thread <-> channel d, 16 states in regs.
// Per-timestep B(t)/C(t) vectors staged once through LDS.
// ---------------------------------------------------------------------------
__global__ void mamba_scan(const float* __restrict__ xs_act,
                           const float* __restrict__ delta,
                           const float* __restrict__ xp,     // (B*L,80): [48:64)=B, [64:80)=C
                           const float* __restrict__ A_log,
                           const float* __restrict__ Dp,
                           const float* __restrict__ xz,     // res = cols [DM, 2*DM)
                           float* __restrict__ y)
{
    const int d = blockIdx.x * blockDim.x + threadIdx.x;
    const int b = blockIdx.y;

    float Ac[NS];
#pragma unroll
    for (int n = 0; n < NS; ++n) Ac[n] = -__expf(A_log[d * NS + n]);
    const float Dd = Dp[d];

    float st[NS];
#pragma unroll
    for (int n = 0; n < NS; ++n) st[n] = 0.0f;

    __shared__ float sB[NS];
    __shared__ float sC[NS];

    for (int t = 0; t < SEQ_L; ++t) {
        const long row = (long)(b * SEQ_L + t);
        if (threadIdx.x < 2 * NS) {
            const int n = threadIdx.x & (NS - 1);
            const float v = xp[row * XP_W + DR + (threadIdx.x >> 4) * NS + n];
            if (threadIdx.x < NS) sB[n] = v; else sC[n] = v;
        }
        __syncthreads();

        const float dt = delta[row * DM + d];
        const float u  = xs_act[row * DM + d];
        const float du = dt * u;
        float acc = 0.0f;
#pragma unroll
        for (int n = 0; n < NS; ++n) {
            st[n] = __expf(dt * Ac[n]) * st[n] + du * sB[n];
            acc += st[n] * sC[n];
        }
        const float r  = xz[row * (2 * DM) + DM + d];
        const float sr = r / (1.0f + __expf(-r));
        y[row * DM + d] = (acc + u * Dd) * sr;
        __syncthreads();
    }
}

// ---------------------------------------------------------------------------
extern "C" void kernel_launch(void* const* d_in, const int* in_sizes, int n_in,
                              void* d_out, int out_size, void* d_ws, size_t ws_size,
                              hipStream_t stream)
{
    const float* x       = (const float*)d_in[0];
    const float* W_in    = (const float*)d_in[1];
    const float* conv_w  = (const float*)d_in[2];
    const float* conv_b  = (const float*)d_in[3];
    const float* W_x     = (const float*)d_in[4];
    const float* W_delta = (const float*)d_in[5];
    const float* b_delta = (const float*)d_in[6];
    const float* A_log   = (const float*)d_in[7];
    const float* D_par   = (const float*)d_in[8];
    const float* W_out   = (const float*)d_in[9];
    float* out = (float*)d_out;

    const int BL = B_SZ * SEQ_L;                 // 2048

    // Workspace layout (fp32 elements), total ~32.1 MB
    float* ws     = (float*)d_ws;
    float* xz     = ws;                          // BL * 1536
    float* xs_act = xz     + (long)BL * 2 * DM;  // BL * 768
    float* xp     = xs_act + (long)BL * DM;      // BL * 80
    float* delta  = xp     + (long)BL * XP_W;    // BL * 768
    float* yv     = delta  + (long)BL * DM;      // BL * 768

    auto gemm_tiled = [&](const float* Am, int lda, const float* Bm, int ldb,
                          float* Cm, int ldc, int M, int N, int K,
                          const float* bias, int ep) {
        const int blocks = (M / BM) * (N / BN);
        mamba_wmma_gemm_tiled<<<blocks, 256, 0, stream>>>(Am, lda, Bm, ldb, Cm, ldc,
                                                          M, N, K, bias, ep);
    };

    // 1) input projection: xz = x @ W_in        (2048 x 1536, K=768), tiled
    gemm_tiled(x, DM, W_in, 2 * DM, xz, 2 * DM, BL, 2 * DM, DM, nullptr, 0);

    // 2) depthwise causal conv + SiLU
    {
        const int tot = BL * DM;
        mamba_conv_silu<<<(tot + 255) / 256, 256, 0, stream>>>(xz, conv_w, conv_b, xs_act);
    }

    // 3) SSM projection: xp = xs_act @ W_x      (2048 x 80, K=768), direct (N%128 != 0)
    {
        const int waves  = (BL / 16) * (XP_W / 16);
        const int blocks = (waves + 7) / 8;
        mamba_wmma_gemm_f32<<<blocks, 256, 0, stream>>>(xs_act, DM, W_x, XP_W, xp, XP_W,
                                                        BL, XP_W, DM, nullptr, 0);
    }

    // 4) delta = clip(softplus(xp[:, :48] @ W_delta + b_delta))  (2048 x 768, K=48), tiled
    gemm_tiled(xp, XP_W, W_delta, DM, delta, DM, BL, DM, DR, b_delta, 1);

    // 5) selective scan + D skip + silu(res) gate
    mamba_scan<<<dim3(DM / 256, B_SZ), 256, 0, stream>>>(xs_act, delta, xp,
                                                         A_log, D_par, xz, yv);

    // 6) output projection: out = y @ W_out     (2048 x 768, K=768), tiled
    gemm_tiled(yv, DM, W_out, DM, out, DM, BL, DM, DM, nullptr, 0);
}